// BallQuery_47545287967129
// MI455X (gfx1250) — compile-verified
//
#include <hip/hip_runtime.h>

typedef __attribute__((ext_vector_type(2))) float v2f;
typedef __attribute__((ext_vector_type(8))) float v8f;

#define NBATCH 4
#define NQ     2048
#define NK     8192
#define KNN    64
#define RAD2   0.01f   // 0.1^2

// One wave32 = one 16-query tile. WMMA 16x16x4 F32 computes the full 16x16
// squared-distance tile per key chunk:  D = (-2Q) x K^T + (|q|^2 + |k|^2).
// Ordered neighbor append via ballot + popcount prefix (preserves key order).
// Per-row counters / first-hit indices are wave-uniform scalars (SGPRs).
__global__ __launch_bounds__(256) void ballquery_wmma_kernel(
    const float* __restrict__ query, const float* __restrict__ key,
    int* __restrict__ out)
{
    const int lane = threadIdx.x & 31;
    const int wave = threadIdx.x >> 5;
    const int tile = blockIdx.x * 8 + wave;        // 0..511
    const int b    = tile >> 7;                    // tile / 128
    const int q0   = (tile & 127) << 4;            // *16
    const int col  = lane & 15;
    const bool hi  = lane >= 16;

    const float* qb = query + (size_t)b * NQ * 3;
    const float* kb = key   + (size_t)b * NK * 3;

    // A matrix (16x4, 32-bit layout): row M = lane%16.
    // lanes 0-15 hold K=0,1 (-2x,-2y); lanes 16-31 hold K=2,3 (-2z, 0).
    const float qx = qb[(q0 + col) * 3 + 0];
    const float qy = qb[(q0 + col) * 3 + 1];
    const float qz = qb[(q0 + col) * 3 + 2];
    v2f a;
    a.x = hi ? (-2.0f * qz) : (-2.0f * qx);
    a.y = hi ? 0.0f         : (-2.0f * qy);

    // |q|^2 per accumulator slice: slice v covers row v (lanes<16) and v+8 (lanes>=16)
    float qn[8];
#pragma unroll
    for (int v = 0; v < 8; ++v) {
        const int row = q0 + v + (hi ? 8 : 0);
        const float x = qb[row * 3 + 0], y = qb[row * 3 + 1], z = qb[row * 3 + 2];
        qn[v] = x * x + y * y + z * z;
    }

    // Wave-uniform per-row state (fully unrolled -> scalar registers).
    int cnt[16], fst[16];
#pragma unroll
    for (int r = 0; r < 16; ++r) { cnt[r] = 0; fst[r] = 0; }

    const size_t outbase = ((size_t)b * NQ + q0) * KNN;

    // Software-pipelined key fetch: preload chunk 0.
    float kx, ky, kz;
    {
        const int n0 = col;
        kx = kb[n0 * 3 + 0]; ky = kb[n0 * 3 + 1]; kz = kb[n0 * 3 + 2];
    }

    for (int kb0 = 0; kb0 < NK; kb0 += 16) {
        const float cx = kx, cy = ky, cz = kz;
        // Prefetch next chunk (wraps to chunk 0 on the last iteration).
        const int np = ((kb0 + 16) & (NK - 1)) + col;
        kx = kb[np * 3 + 0]; ky = kb[np * 3 + 1]; kz = kb[np * 3 + 2];

        const float kn = cx * cx + cy * cy + cz * cz;

        // B matrix (4x16): column N = lane%16; lanes<16 K=0,1; lanes>=16 K=2,3.
        v2f bm;
        bm.x = hi ? cz   : cx;
        bm.y = hi ? 0.0f : cy;

        // Fold |q|^2 + |k|^2 into the accumulator -> WMMA emits d^2 directly.
        v8f c;
#pragma unroll
        for (int v = 0; v < 8; ++v) c[v] = qn[v] + kn;

        v8f d2 = __builtin_amdgcn_wmma_f32_16x16x4_f32(
            false, a, false, bm, (short)0, c, false, false);

#pragma unroll
        for (int v = 0; v < 8; ++v) {
            const bool within = d2[v] < RAD2;           // lane's own tile bit
            const unsigned ball = (unsigned)__ballot(within); // wave32: low 32 bits
            if (!ball) continue;                        // wave-uniform skip (~87%)
            const unsigned m0 = ball & 0xffffu;         // row v
            const unsigned m1 = ball >> 16;             // row v+8

            // Slot = row count + rank among set bits below my column.
            const int      myc = hi ? cnt[v + 8] : cnt[v];
            const unsigned mym = hi ? m1 : m0;
            const int pos = myc + __popc(mym & ((1u << col) - 1u));
            if (within && pos < KNN) {
                const int rowoff = (hi ? (v + 8) : v) * KNN;
                out[outbase + rowoff + pos] = kb0 + col;
            }

            // Branchless wave-uniform counter / first-hit updates (SALU).
            const int f0 = __ffs((int)m0) - 1;
            const int f1 = __ffs((int)m1) - 1;
            fst[v]     = (cnt[v]     == 0 && m0) ? (kb0 + f0) : fst[v];
            fst[v + 8] = (cnt[v + 8] == 0 && m1) ? (kb0 + f1) : fst[v + 8];
            cnt[v]     += __popc(m0);
            cnt[v + 8] += __popc(m1);
        }

        // Periodic early exit: all 16 rows saturated (checked every 8 chunks).
        if ((kb0 & 127) == 112) {
            bool done = true;
#pragma unroll
            for (int r = 0; r < 16; ++r) done = done && (cnt[r] >= KNN);
            if (done) break;
        }
    }

    // Pad unfilled slots with the first in-range index (0 if none).
#pragma unroll
    for (int r = 0; r < 16; ++r) {
        int c = cnt[r];
        if (c > KNN) c = KNN;
        const int f = fst[r];
        const size_t rowb = outbase + (size_t)r * KNN;
        if (lane >= c)      out[rowb + lane]      = f;
        if (lane + 32 >= c) out[rowb + lane + 32] = f;
    }
}

extern "C" void kernel_launch(void* const* d_in, const int* in_sizes, int n_in,
                              void* d_out, int out_size, void* d_ws, size_t ws_size,
                              hipStream_t stream) {
    const float* query = (const float*)d_in[0];   // (4, 2048, 3) f32
    const float* key   = (const float*)d_in[1];   // (4, 8192, 3) f32
    int* out = (int*)d_out;                       // (4, 2048, 64) i32

    // 512 query tiles (4 * 2048/16), 8 waves per 256-thread block -> 64 blocks
    hipLaunchKernelGGL(ballquery_wmma_kernel, dim3(64), dim3(256), 0, stream,
                       query, key, out);
}